// Hd_45397804319098
// MI455X (gfx1250) — compile-verified
//
#include <hip/hip_runtime.h>
#include <math.h>

// Problem constants (fixed by the reference setup).
#define B_   8
#define N1_  512
#define N2_  512
#define D_   128
#define TILES_I (N1_/16)                 // 32
#define TILES_J (N2_/16)                 // 32
#define TOTAL_TILES (B_*TILES_I*TILES_J) // 8192

typedef __attribute__((ext_vector_type(2))) float v2f;
typedef __attribute__((ext_vector_type(8))) float v8f;

// ---------------------------------------------------------------------------
// Kernel 1: per-row squared norms + init min-buffers to +inf.
// idx in [0, B*N1): v1 row norms -> n1, bm2sq=inf
// idx in [B*N1, B*N1+B*N2): v2 row norms -> n2, bm1sq=inf
// ---------------------------------------------------------------------------
__global__ __launch_bounds__(256) void hd_init(const float* __restrict__ v1,
                                               const float* __restrict__ v2,
                                               float* __restrict__ n1,
                                               float* __restrict__ n2,
                                               float* __restrict__ bm2sq,
                                               float* __restrict__ bm1sq) {
    int idx = blockIdx.x * blockDim.x + threadIdx.x;
    if (idx < B_ * N1_) {
        const float4* row = (const float4*)(v1 + (size_t)idx * D_);
        float s = 0.f;
#pragma unroll
        for (int kk = 0; kk < D_ / 4; ++kk) {
            float4 f = row[kk];
            s += f.x * f.x + f.y * f.y + f.z * f.z + f.w * f.w;
        }
        n1[idx]    = s;
        bm2sq[idx] = INFINITY;
    } else if (idx < B_ * N1_ + B_ * N2_) {
        int id2 = idx - B_ * N1_;
        const float4* row = (const float4*)(v2 + (size_t)id2 * D_);
        float s = 0.f;
#pragma unroll
        for (int kk = 0; kk < D_ / 4; ++kk) {
            float4 f = row[kk];
            s += f.x * f.x + f.y * f.y + f.z * f.z + f.w * f.w;
        }
        n2[id2]    = s;
        bm1sq[id2] = INFINITY;
    }
}

// ---------------------------------------------------------------------------
// Kernel 2: one wave32 per 16x16 distance tile.
// G = A x B with A = v2 tile (16xK), B = v1^T tile (Kx16), via
// V_WMMA_F32_16X16X4_F32 chained over K=128 (32 WMMAs).
// ISA f32 16x4 A layout: lane lr in 0..15 holds row M=lr; half-wave hi=0 has
// K={0,1} in (VGPR0,VGPR1), hi=1 has K={2,3}.  B mirrors this with N=lr.
// Both reduce to a single float2 load at column offset k+2*hi.
// C/D layout: acc[r] of lane (lr,hi) holds D[M = r+8*hi, N = lr].
// ---------------------------------------------------------------------------
__global__ __launch_bounds__(256) void hd_tiles(const float* __restrict__ v1,
                                                const float* __restrict__ v2,
                                                const int*   __restrict__ sz1,
                                                const int*   __restrict__ sz2,
                                                const float* __restrict__ n1,
                                                const float* __restrict__ n2,
                                                unsigned*    __restrict__ bm2u,
                                                unsigned*    __restrict__ bm1u) {
    const int wv   = blockIdx.x * (blockDim.x >> 5) + (threadIdx.x >> 5);
    const int lane = threadIdx.x & 31;
    const int b    = wv >> 10;                 // 1024 tiles per batch
    const int t    = wv & 1023;
    const int j0   = (t >> 5) << 4;            // v2 tile origin (M axis)
    const int i0   = (t & 31) << 4;            // v1 tile origin (N axis)
    const int lr   = lane & 15;
    const int hi   = lane >> 4;

    const float* arow = v2 + ((size_t)(b * N2_) + j0 + lr) * D_ + 2 * hi;
    const float* brow = v1 + ((size_t)(b * N1_) + i0 + lr) * D_ + 2 * hi;

    v8f acc = {0.f, 0.f, 0.f, 0.f, 0.f, 0.f, 0.f, 0.f};
#pragma unroll
    for (int k = 0; k < D_; k += 4) {
        v2f a  = *(const v2f*)(arow + k);
        v2f bb = *(const v2f*)(brow + k);
        // 8 args: (neg_a, A, neg_b, B, c_mod, C, reuse_a, reuse_b)
        acc = __builtin_amdgcn_wmma_f32_16x16x4_f32(
            false, a, false, bb, (short)0, acc, false, false);
    }

    const int  s1 = sz1[b];
    const int  s2 = sz2[b];
    const float ni = n1[b * N1_ + i0 + lr];     // ||v1_i||^2, i = i0+lr
    const bool vi  = (i0 + lr) < s1;

    float vals[8];
    float vmin = INFINITY;
#pragma unroll
    for (int r = 0; r < 8; ++r) {
        const int   j  = j0 + 8 * hi + r;       // v2 row of this acc element
        float d2 = n2[b * N2_ + j] + ni - 2.f * acc[r];
        d2 = fmaxf(d2, 0.f);                    // clamp fp rounding negatives
        const float v = (vi && (j < s2)) ? d2 : INFINITY;
        vals[r] = v;
        vmin = fminf(vmin, v);
    }

    // bm2[i] = min over j: lane-local min over 8 j's, then combine halves.
    vmin = fminf(vmin, __shfl_xor(vmin, 16, 32));
    if (lane < 16)
        atomicMin(&bm2u[b * N1_ + i0 + lr], __float_as_uint(vmin));

    // bm1[j] = min over i: reduce across the 16 lanes of each half-wave.
#pragma unroll
    for (int r = 0; r < 8; ++r) {
        float w = vals[r];
        w = fminf(w, __shfl_xor(w, 1, 32));
        w = fminf(w, __shfl_xor(w, 2, 32));
        w = fminf(w, __shfl_xor(w, 4, 32));
        w = fminf(w, __shfl_xor(w, 8, 32));
        if (lr == 0)
            atomicMin(&bm1u[b * N2_ + j0 + 8 * hi + r], __float_as_uint(w));
    }
}

// ---------------------------------------------------------------------------
// Kernel 3: per-batch masked max of the two min-arrays, sqrt once.
// ---------------------------------------------------------------------------
__global__ __launch_bounds__(256) void hd_finalize(const float* __restrict__ bm1sq,
                                                   const float* __restrict__ bm2sq,
                                                   const int*   __restrict__ sz1,
                                                   const int*   __restrict__ sz2,
                                                   float*       __restrict__ out) {
    __shared__ float red[256];
    const int b  = blockIdx.x;
    const int s1 = sz1[b];
    const int s2 = sz2[b];
    float m = 0.f;
    if (s1 > 0 && s2 > 0) {
        for (int j = threadIdx.x; j < s2; j += blockDim.x)
            m = fmaxf(m, bm1sq[b * N2_ + j]);
        for (int i = threadIdx.x; i < s1; i += blockDim.x)
            m = fmaxf(m, bm2sq[b * N1_ + i]);
    }
    red[threadIdx.x] = m;
    __syncthreads();
    for (int s = 128; s > 0; s >>= 1) {
        if (threadIdx.x < s)
            red[threadIdx.x] = fmaxf(red[threadIdx.x], red[threadIdx.x + s]);
        __syncthreads();
    }
    if (threadIdx.x == 0)
        out[b] = sqrtf(red[0]);   // sqrt(max(d^2)) == max(sqrt(d^2))
}

// ---------------------------------------------------------------------------
// Launch. Input order: v1, am1(unused), sz1, v2, am2(unused), sz2.
// Workspace (floats): [bm2sq: B*N1][bm1sq: B*N2][n1: B*N1][n2: B*N2] = 64 KB.
// ---------------------------------------------------------------------------
extern "C" void kernel_launch(void* const* d_in, const int* in_sizes, int n_in,
                              void* d_out, int out_size, void* d_ws, size_t ws_size,
                              hipStream_t stream) {
    const float* v1  = (const float*)d_in[0];
    const int*   sz1 = (const int*)  d_in[2];
    const float* v2  = (const float*)d_in[3];
    const int*   sz2 = (const int*)  d_in[5];
    float*       out = (float*)d_out;

    float* ws    = (float*)d_ws;
    float* bm2sq = ws;                       // B*N1
    float* bm1sq = ws + B_ * N1_;            // B*N2
    float* n1    = ws + B_ * N1_ + B_ * N2_; // B*N1
    float* n2    = n1 + B_ * N1_;            // B*N2

    const int rows = B_ * (N1_ + N2_);       // 8192
    hd_init<<<(rows + 255) / 256, 256, 0, stream>>>(v1, v2, n1, n2, bm2sq, bm1sq);

    hd_tiles<<<TOTAL_TILES / 8, 256, 0, stream>>>(
        v1, v2, sz1, sz2, n1, n2, (unsigned*)bm2sq, (unsigned*)bm1sq);

    hd_finalize<<<B_, 256, 0, stream>>>(bm1sq, bm2sq, sz1, sz2, out);
}